// Simple_Concat_Layer_59055800320842
// MI455X (gfx1250) — compile-verified
//
#include <hip/hip_runtime.h>
#include <hip/hip_bf16.h>

// Problem dims (fixed by the reference)
#define DIM_B 4
#define DIM_C 16
#define DIM_L 512
#define DIM_H 1024
// rows per tensor = B*C*L = 32768 ; row-tiles per tensor = 2048

typedef __attribute__((ext_vector_type(2))) float v2f;
typedef __attribute__((ext_vector_type(4))) float v4f;
typedef __attribute__((ext_vector_type(8))) float v8f;

// ---------------------------------------------------------------------------
// Kernel 1: GEMV reductions via V_WMMA_F32_16X16X4_F32.
// One wave32 owns one 16-row tile of either start (waves [0,2048)) or end
// (waves [2048,4096)). A-layout (fp32 16x4): lanes 0-15 hold K=0,1 in the two
// A VGPRs, lanes 16-31 hold K=2,3 — i.e. lane loads float2 at col k0+2*(lane>>4)
// of its row. B is the matching v-chunk replicated across all 16 N columns,
// so every column of the 16x16 f32 accumulator equals the exact dot product.
// Results written transposed: dst[(b*L + l)*16 + c].
// ---------------------------------------------------------------------------
__global__ __launch_bounds__(256)
void concat_score_reduce_wmma(const float* __restrict__ start_h,
                              const float* __restrict__ end_h,
                              const float* __restrict__ v,
                              float* __restrict__ s_t,   // [B*L*16]
                              float* __restrict__ e_t) { // [B*L*16]
    const int wid  = (blockIdx.x * 256 + threadIdx.x) >> 5;  // 0..4095
    const int lane = threadIdx.x & 31;
    const int half = lane >> 4;    // 0: K=0,1   1: K=2,3
    const int mrow = lane & 15;    // row within tile

    const bool is_end = wid >= 2048;
    const int  tile   = is_end ? (wid - 2048) : wid;        // 0..2047
    const float* __restrict__ src  = is_end ? end_h : start_h;
    const float* __restrict__ vseg = v + (is_end ? DIM_H : 0);
    float* __restrict__ dst        = is_end ? e_t : s_t;

    const int row0 = tile * 16;                              // first global row
    const float* __restrict__ rowp = src + (size_t)(row0 + mrow) * DIM_H + 2 * half;
    const float* __restrict__ vp   = vseg + 2 * half;

    v8f c0 = {};  // even K-chunks
    v8f c1 = {};  // odd K-chunks (breaks the WMMA accumulation RAW chain)

#pragma unroll 4
    for (int k0 = 0; k0 < DIM_H; k0 += 8) {
        v2f a0 = *(const v2f*)(rowp + k0);
        v2f b0 = *(const v2f*)(vp   + k0);
        v2f a1 = *(const v2f*)(rowp + k0 + 4);
        v2f b1 = *(const v2f*)(vp   + k0 + 4);
        // 8 args: (neg_a, A, neg_b, B, c_mod, C, reuse_a, reuse_b)
        c0 = __builtin_amdgcn_wmma_f32_16x16x4_f32(false, a0, false, b0,
                                                   (short)0, c0, false, false);
        c1 = __builtin_amdgcn_wmma_f32_16x16x4_f32(false, a1, false, b1,
                                                   (short)0, c1, false, false);
    }

    // C/D layout: VGPR g -> lanes 0-15: M=g (N=lane), lanes 16-31: M=g+8.
    // All N columns are identical, so lane 0 holds rows 0..7 in c[0..7] and
    // lane 16 holds rows 8..15. Write transposed to [b, l, c].
    if (mrow == 0) {
#pragma unroll
        for (int g = 0; g < 8; ++g) {
            int r  = row0 + half * 8 + g;          // global row = (b*16+c)*512 + l
            int bb = r >> 13;                      // / (C*L)
            int cc = (r >> 9) & 15;                // / L  % C
            int ll = r & 511;                      // % L
            dst[(((bb << 9) + ll) << 4) + cc] = c0[g] + c1[g];
        }
    }
}

// ---------------------------------------------------------------------------
// Kernel 2: out[b,i,j,c] = s_t[b,i,c] + e_t[b,j,c], one float4 per thread.
// Pure streaming-write kernel; s_t/e_t (256 KB) live in cache. NT stores.
// ---------------------------------------------------------------------------
__global__ __launch_bounds__(256)
void concat_score_fill(const float* __restrict__ s_t,
                       const float* __restrict__ e_t,
                       float* __restrict__ out) {
    const int g = blockIdx.x * 256 + threadIdx.x;   // float4 index, 0..4194303
    const int c4 = g & 3;
    const int j  = (g >> 2)  & 511;
    const int i  = (g >> 11) & 511;
    const int b  = g >> 20;

    const v4f* __restrict__ sp = (const v4f*)s_t;
    const v4f* __restrict__ ep = (const v4f*)e_t;

    v4f s4 = sp[(((b << 9) + i) << 2) + c4];
    v4f e4 = ep[(((b << 9) + j) << 2) + c4];
    v4f o  = s4 + e4;

    __builtin_nontemporal_store(o, ((v4f*)out) + g);
}

extern "C" void kernel_launch(void* const* d_in, const int* in_sizes, int n_in,
                              void* d_out, int out_size, void* d_ws, size_t ws_size,
                              hipStream_t stream) {
    const float* start_h = (const float*)d_in[0];  // [B,C,L,H]
    const float* end_h   = (const float*)d_in[1];  // [B,C,L,H]
    const float* v       = (const float*)d_in[2];  // [2H]
    float* out = (float*)d_out;                    // [B,L,L,C]

    // Workspace: s_t and e_t, each B*L*16 floats (128 KB each)
    float* s_t = (float*)d_ws;
    float* e_t = s_t + DIM_B * DIM_L * DIM_C;

    // Phase 1: 4096 waves (2048 tiles per tensor), 8 waves per 256-thread block
    concat_score_reduce_wmma<<<512, 256, 0, stream>>>(start_h, end_h, v, s_t, e_t);

    // Phase 2: 16,777,216 out floats / 4 per thread / 256 per block
    concat_score_fill<<<16384, 256, 0, stream>>>(s_t, e_t, out);
}